// SpikingConv2DLayer_4200478015878
// MI455X (gfx1250) — compile-verified
//
#include <hip/hip_runtime.h>

// ---------------------------------------------------------------------------
// SpikingConv2DLayer for MI455X (gfx1250), wave32, WMMA f32 16x16x4 +
// gfx1250 async global<->LDS staging (ASYNCcnt).
//
// Shapes: x(32,32,100,128) f32, w(128,32,3,3) f32, beta(1), b(128)
// out: spikes (32,128,100,128) f32.  conv_x has the same shape as out, so the
// conv writes into d_out and the temporal scan runs IN PLACE.
// ---------------------------------------------------------------------------

typedef __attribute__((ext_vector_type(2))) float v2f;
typedef __attribute__((ext_vector_type(8))) float v8f;
typedef __attribute__((ext_vector_type(2))) int   i32x2;
typedef __attribute__((ext_vector_type(4))) int   i32x4;

#define WMMA_F32X4(A, B, C) \
    __builtin_amdgcn_wmma_f32_16x16x4_f32(false, (A), false, (B), (short)0, (C), false, false)

#define NB   32
#define CIN  32
#define NT   100
#define NW   128
#define NCO  128
#define KCONV 288
#define EPSF 1e-8f

// ------------------- gfx1250 async global<->LDS helpers --------------------
#if __has_builtin(__builtin_amdgcn_global_load_async_to_lds_b64) &&  \
    __has_builtin(__builtin_amdgcn_global_load_async_to_lds_b128) && \
    __has_builtin(__builtin_amdgcn_global_store_async_from_lds_b32) && \
    __has_builtin(__builtin_amdgcn_global_store_async_from_lds_b128) && \
    __has_builtin(__builtin_amdgcn_s_wait_asynccnt)
#define HAVE_ASYNC 1
#else
#define HAVE_ASYNC 0
#endif

#if HAVE_ASYNC
typedef __attribute__((address_space(1))) int*   g32p;
typedef __attribute__((address_space(1))) i32x2* g64p;
typedef __attribute__((address_space(1))) i32x4* g128p;
typedef __attribute__((address_space(3))) int*   l32p;
typedef __attribute__((address_space(3))) i32x2* l64p;
typedef __attribute__((address_space(3))) i32x4* l128p;

__device__ __forceinline__ void a_ld64(const float* g, float* l) {
    __builtin_amdgcn_global_load_async_to_lds_b64((g64p)g, (l64p)l, 0, 0);
}
__device__ __forceinline__ void a_ld128(const float* g, float* l) {
    __builtin_amdgcn_global_load_async_to_lds_b128((g128p)g, (l128p)l, 0, 0);
}
__device__ __forceinline__ void a_st32(float* g, const float* l) {
    __builtin_amdgcn_global_store_async_from_lds_b32((g32p)g, (l32p)l, 0, 0);
}
__device__ __forceinline__ void a_st128(float* g, const float* l) {
    __builtin_amdgcn_global_store_async_from_lds_b128((g128p)g, (l128p)l, 0, 0);
}
__device__ __forceinline__ void a_wait() { __builtin_amdgcn_s_wait_asynccnt(0); }
#else
__device__ __forceinline__ void a_ld64(const float* g, float* l) {
    l[0] = g[0]; l[1] = g[1];
}
__device__ __forceinline__ void a_ld128(const float* g, float* l) {
    *(float4*)l = *(const float4*)g;
}
__device__ __forceinline__ void a_st32(float* g, const float* l) { g[0] = l[0]; }
__device__ __forceinline__ void a_st128(float* g, const float* l) {
    *(float4*)g = *(const float4*)l;
}
__device__ __forceinline__ void a_wait() {}
#endif

// ---------------------------------------------------------------------------
// Kernel 0a: Gram matrix d[i][j] = <w_i, w_j> (128x128, K=288) + inv_norm.
// ---------------------------------------------------------------------------
__global__ __launch_bounds__(128) void gram_kernel(const float* __restrict__ wgt,
                                                   float* __restrict__ dmat,
                                                   float* __restrict__ invn) {
    __shared__ float wi[KCONV];
    const int i   = blockIdx.x;
    const int tid = threadIdx.x;
    for (int k = tid; k < KCONV; k += 128) wi[k] = wgt[i * KCONV + k];
    __syncthreads();
    const float* wj = wgt + tid * KCONV;
    float acc = 0.f;
#pragma unroll 8
    for (int k = 0; k < KCONV; ++k) acc += wi[k] * wj[k];
    dmat[i * NCO + tid] = acc;
    if (tid == i) invn[i] = 1.0f / (acc + EPSF);
}

// ---------------------------------------------------------------------------
// Kernel 0b: weights -> K-padded implicit-GEMM layout:
//   wT[(ci*3+kh)*512 + kws*128 + co] = (kws<3) ? w[co,ci,kh,kws] : 0
// ---------------------------------------------------------------------------
__global__ __launch_bounds__(256) void wtrans_kernel(const float* __restrict__ wgt,
                                                     float* __restrict__ wT) {
    int e = blockIdx.x * 256 + threadIdx.x;
    if (e >= 96 * 4 * NCO) return;
    int co   = e & 127;
    int kws  = (e >> 7) & 3;
    int cikh = e >> 9;
    int ci   = cikh / 3;
    int kh   = cikh % 3;
    wT[e] = (kws < 3) ? wgt[((co * CIN + ci) * 3 + kh) * 3 + kws] : 0.f;
}

// ---------------------------------------------------------------------------
// Kernel 1: 3x3 conv as implicit GEMM, V_WMMA_F32_16X16X4_F32.
// Block per (b,t): M=128 (all w) x N=128 (all co), K=384 (padded).
// 24 rounds x 4 staged (ci,kh) slices -> 32 WMMAs/wave between barriers.
// Weight slices staged with async b128 loads; output tile transposed through
// LDS and written with async b128 stores.
// ---------------------------------------------------------------------------
#define WSLS 144   // padded row stride (dwords) for one staged weight K-row

__global__ __launch_bounds__(256) void conv_kernel(const float* __restrict__ x,
                                                   const float* __restrict__ wT,
                                                   float* __restrict__ out) {
    const int bt   = blockIdx.x;
    const int b    = bt / NT;
    const int t    = bt % NT;
    const int tid  = threadIdx.x;
    const int lane = tid & 31;
    const int wid  = tid >> 5;
    const int mbase = (wid & 3) * 32;
    const int nbase = (wid >> 2) * 64;
    const int h    = lane >> 4;
    const int l15  = lane & 15;

    __shared__ float xrow4[4 * 132];        // 4 slices of x row (-1..128 + pad)
    __shared__ float wsl[4 * 4 * WSLS];     // 4 slices x [kws][co(padded)]
    __shared__ float otile[64 * 132];       // output transpose buffer

    v8f acc[2][4];
#pragma unroll
    for (int mf = 0; mf < 2; ++mf)
#pragma unroll
        for (int nf = 0; nf < 4; ++nf) acc[mf][nf] = (v8f){0,0,0,0,0,0,0,0};

    int ci = 0, kh = 0;                     // running cikh = 4*r + sl
    for (int r = 0; r < 24; ++r) {
        __syncthreads();                    // protect prior round's frag reads

        // async-stage 4 weight slices: 2048 floats = 512 x b128, 2 per thread
#pragma unroll
        for (int i = 0; i < 2; ++i) {
            int e   = tid + 256 * i;        // 0..511
            int sl  = e >> 7;
            int kws = (e >> 5) & 3;
            int c4  = e & 31;
            a_ld128(wT + (size_t)(4 * r + sl) * 512 + kws * 128 + c4 * 4,
                    &wsl[sl * 4 * WSLS + kws * WSLS + c4 * 4]);
        }
        // sync-stage 4 x rows (needs zero-fill at T/W borders)
#pragma unroll
        for (int sl = 0; sl < 4; ++sl) {
            int th = t + kh - 1;
            if (tid < 132) {
                int wm = tid - 1;
                float v = 0.f;
                if (th >= 0 && th < NT && wm >= 0 && wm < NW)
                    v = x[(((size_t)b * CIN + ci) * NT + th) * NW + wm];
                xrow4[sl * 132 + tid] = v;
            }
            kh++; if (kh == 3) { kh = 0; ci++; }
        }
        a_wait();
        __syncthreads();

#pragma unroll
        for (int sl = 0; sl < 4; ++sl) {
            v2f afr[2], bfr[4];
#pragma unroll
            for (int mf = 0; mf < 2; ++mf) {
                int wbar = mbase + mf * 16 + l15;
                afr[mf].x = xrow4[sl * 132 + wbar + 2 * h];
                afr[mf].y = xrow4[sl * 132 + wbar + 2 * h + 1];
            }
#pragma unroll
            for (int nf = 0; nf < 4; ++nf) {
                int co = nbase + nf * 16 + l15;
                bfr[nf].x = wsl[sl * 4 * WSLS + (2 * h) * WSLS + co];
                bfr[nf].y = wsl[sl * 4 * WSLS + (2 * h + 1) * WSLS + co];
            }
#pragma unroll
            for (int mf = 0; mf < 2; ++mf)
#pragma unroll
                for (int nf = 0; nf < 4; ++nf)
                    acc[mf][nf] = WMMA_F32X4(afr[mf], bfr[nf], acc[mf][nf]);
        }
    }

    // Transpose D tiles through LDS (2 chunks of 64 co rows) -> async stores.
    for (int chunk = 0; chunk < 2; ++chunk) {
        __syncthreads();
        a_wait();                           // chunk0 stores done before reuse
        if ((wid >> 2) == chunk) {
#pragma unroll
            for (int mf = 0; mf < 2; ++mf)
#pragma unroll
                for (int nf = 0; nf < 4; ++nf)
#pragma unroll
                    for (int rr = 0; rr < 8; ++rr) {
                        int wbar = mbase + mf * 16 + rr + 8 * h;
                        int col  = nf * 16 + l15;
                        otile[col * 132 + wbar] = acc[mf][nf][rr];
                    }
        }
        __syncthreads();
        for (int i = 0; i < 8; ++i) {
            int e   = tid + 256 * i;        // 0..2047
            int cor = e >> 5;
            int w4  = e & 31;
            int co  = chunk * 64 + cor;
            a_st128(&out[(((size_t)b * NCO + co) * NT + t) * NW + w4 * 4],
                    &otile[cor * 132 + w4 * 4]);
        }
    }
}

// ---------------------------------------------------------------------------
// Kernel 2: temporal scan, IN PLACE on d_out.
// Block = one b, 16 consecutive w, all 128 co; 8 waves of one 16-co slice.
// Per step: rst = spk(16x128) @ d(128x128) via 32 WMMAs/wave (d fragments
// preloaded in VGPRs); leaky-integrate + threshold in the WMMA C/D layout.
// Input slice staged with async b64 loads; spikes written back with async
// b32 stores from LDS.
// ---------------------------------------------------------------------------
#define SPKS 130
#define INPS 18

__global__ __launch_bounds__(256) void scan_kernel(const float* __restrict__ dmat,
                                                   const float* __restrict__ invn,
                                                   const float* __restrict__ bias,
                                                   const float* __restrict__ beta_p,
                                                   float* __restrict__ io) {
    const int blk  = blockIdx.x;
    const int b    = blk >> 3;
    const int w0   = (blk & 7) * 16;
    const int tid  = threadIdx.x;
    const int lane = tid & 31;
    const int wid  = tid >> 5;
    const int nb   = wid * 16;
    const int h    = lane >> 4;
    const int l15  = lane & 15;

    __shared__ float spk_lds[16 * SPKS];   // [seq][co]
    __shared__ float inp_lds[NCO * INPS];  // [co][seq]

    for (int e = tid; e < 16 * SPKS; e += 256) spk_lds[e] = 0.f;

    v2f bfr[32];
#pragma unroll
    for (int ks = 0; ks < 32; ++ks) {
        int k  = ks * 4 + 2 * h;
        int co = nb + l15;
        bfr[ks].x = dmat[k * NCO + co];
        bfr[ks].y = dmat[(k + 1) * NCO + co];
    }
    const float inv_n  = invn[nb + l15];
    const float bv     = bias[nb + l15];
    const float beta   = beta_p[0];
    const float ombeta = 1.0f - beta;

    v8f mem = (v8f){0,0,0,0,0,0,0,0};
    const size_t boff = (size_t)b * NCO * NT * NW;
    __syncthreads();

    for (int t = 0; t < NT; ++t) {
        // async load of conv slice (128 co x 16 w): b64, 4 per thread
        for (int i = 0; i < 4; ++i) {
            int e  = tid + 256 * i;         // 0..1023
            int co = e >> 3;
            int s2 = (e & 7) * 2;
            a_ld64(&io[boff + (size_t)co * (NT * NW) + (size_t)t * NW + w0 + s2],
                   &inp_lds[co * INPS + s2]);
        }
        a_wait();                           // loads done; prior spikes stored
        __syncthreads();

        v8f rst = (v8f){0,0,0,0,0,0,0,0};
#pragma unroll
        for (int ks = 0; ks < 32; ++ks) {
            int kk = ks * 4 + 2 * h;
            v2f a;
            a.x = spk_lds[l15 * SPKS + kk];
            a.y = spk_lds[l15 * SPKS + kk + 1];
            rst = WMMA_F32X4(a, bfr[ks], rst);
        }

        float spk_new[8];
#pragma unroll
        for (int rr = 0; rr < 8; ++rr) {
            int s = rr + 8 * h;
            float inp = inp_lds[(nb + l15) * INPS + s];
            float m = (mem[rr] - rst[rr]) * beta + inp * ombeta;
            mem[rr] = m;
            spk_new[rr] = (m * inv_n - bv) > 0.f ? 1.0f : 0.0f;
        }
        __syncthreads();                    // all A-reads done before overwrite
#pragma unroll
        for (int rr = 0; rr < 8; ++rr)
            spk_lds[(rr + 8 * h) * SPKS + (nb + l15)] = spk_new[rr];
        __syncthreads();

        // async in-place spike writeback (overwrites conv slice t)
        for (int i = 0; i < 8; ++i) {
            int e  = tid + 256 * i;
            int co = e >> 4;
            int s  = e & 15;
            a_st32(&io[boff + (size_t)co * (NT * NW) + (size_t)t * NW + w0 + s],
                   &spk_lds[s * SPKS + co]);
        }
    }
}

// ---------------------------------------------------------------------------
extern "C" void kernel_launch(void* const* d_in, const int* in_sizes, int n_in,
                              void* d_out, int out_size, void* d_ws, size_t ws_size,
                              hipStream_t stream) {
    (void)in_sizes; (void)n_in; (void)out_size; (void)ws_size;
    const float* x      = (const float*)d_in[0];
    const float* wgt    = (const float*)d_in[1];
    const float* beta_p = (const float*)d_in[2];
    const float* bias   = (const float*)d_in[3];
    float* out  = (float*)d_out;

    float* dmat = (float*)d_ws;                 // 128*128 floats
    float* invn = dmat + NCO * NCO;             // 128 floats
    float* wT   = invn + NCO;                   // 96*4*128 floats

    gram_kernel<<<NCO, 128, 0, stream>>>(wgt, dmat, invn);
    wtrans_kernel<<<(96 * 4 * NCO + 255) / 256, 256, 0, stream>>>(wgt, wT);
    conv_kernel<<<NB * NT, 256, 0, stream>>>(x, wT, out);
    scan_kernel<<<(NB * NW) / 16, 256, 0, stream>>>(dmat, invn, bias, beta_p, out);
}